// GCN_47614007444004
// MI455X (gfx1250) — compile-verified
//
#include <hip/hip_runtime.h>
#include <hip/hip_bf16.h>

#define N_NODES  50000
#define N_EDGES  800000
#define N_GRAPHS 256
#define IN_C     128
#define HID_C    64
#define OUT_C    2

typedef __attribute__((ext_vector_type(2))) float v2f;
typedef __attribute__((ext_vector_type(8))) float v8f;

// ---------------------------------------------------------------------------
// WMMA GEMM: C[M x 64] = A[M x K] @ B[K x 64], M = 50000 (= 3125 * 16 exact).
// One block = 128 threads = 4 waves. Block owns one 16-row M-tile; wave w owns
// the 16x16 output tile at columns [16w, 16w+16). A-tile staged in LDS with a
// padded stride (K+4) so per-lane fragment reads land on distinct banks.
// Uses full-precision V_WMMA_F32_16X16X4_F32 (reference is f32; GEMM is not
// the bottleneck, so no reason to degrade to bf16).
// ---------------------------------------------------------------------------
template <int K>
__global__ __launch_bounds__(128) void gcn_gemm_wmma(
    const float* __restrict__ A, const float* __restrict__ B,
    float* __restrict__ C, int Mtiles) {
  constexpr int LDA = K + 4;
  __shared__ float tileA[16 * LDA];

  const int mt = blockIdx.x;
  if (mt >= Mtiles) return;                 // uniform per block: EXEC stays all-1s

  const int tid  = threadIdx.x;
  const int wave = tid >> 5;                // 0..3 -> N tile
  const int lane = tid & 31;
  const int half = lane >> 4;               // 0: K+0/K+1, 1: K+2/K+3 (A frag)
  const int l16  = lane & 15;

  // Cooperative, coalesced load of the 16 x K A-tile into LDS.
  for (int f = tid; f < 16 * K; f += 128) {
    const int r = f / K, c = f % K;         // K is compile-time -> shifts
    tileA[r * LDA + c] = A[(size_t)(mt * 16 + r) * K + c];
  }
  __syncthreads();

  const int col = wave * 16 + l16;          // B / C column for this lane

  v8f acc = {};
  #pragma unroll
  for (int k = 0; k < K; k += 4) {
    const int ka = k + half * 2;
    v2f a, b;
    // A 16x4 frag (ISA 7.12.2): lanes 0-15 = rows, v0/v1 = K0/K1; upper half K2/K3
    a.x = tileA[l16 * LDA + ka];
    a.y = tileA[l16 * LDA + ka + 1];
    // B 4x16 frag: lane = column, v0/v1 = rows (K+half*2, K+half*2+1)
    b.x = B[(ka + 0) * HID_C + col];
    b.y = B[(ka + 1) * HID_C + col];
    acc = __builtin_amdgcn_wmma_f32_16x16x4_f32(
        /*neg_a=*/false, a, /*neg_b=*/false, b,
        /*c_mod=*/(short)0, acc, /*reuse_a=*/false, /*reuse_b=*/false);
  }

  // C/D layout: VGPR i -> row (half*8 + i), lane l16 -> column.
  float* Crow = C + (size_t)(mt * 16 + half * 8) * HID_C + col;
  #pragma unroll
  for (int i = 0; i < 8; ++i) Crow[(size_t)i * HID_C] = acc[i];
}

// ---------------------------------------------------------------------------
// Degree with edge weights + self loop (weight 1), then deg -> rsqrt in place.
// ---------------------------------------------------------------------------
__global__ void gcn_deg_init(float* __restrict__ deg) {
  int i = blockIdx.x * blockDim.x + threadIdx.x;
  if (i < N_NODES) deg[i] = 1.0f;           // self-loop contribution
}

__global__ void gcn_deg_accum(const int* __restrict__ dst, const float* __restrict__ ew,
                              float* __restrict__ deg) {
  int e = blockIdx.x * blockDim.x + threadIdx.x;
  if (e < N_EDGES) atomicAdd(&deg[dst[e]], ew[e]);
}

__global__ void gcn_deg_finish(float* __restrict__ deg) {
  int i = blockIdx.x * blockDim.x + threadIdx.x;
  if (i < N_NODES) {
    float d = deg[i];
    deg[i] = (d > 0.0f) ? rsqrtf(d) : 0.0f;
  }
}

// ---------------------------------------------------------------------------
// Edge propagate: 16 lanes per edge, one float4 (16 B) per lane -> full 256 B
// row gathered coalesced, scaled by dinv[s]*ew*dinv[d], scattered with f32
// atomics (global_atomic_add_f32; mostly L2-resident).
// ---------------------------------------------------------------------------
__global__ void gcn_scatter(const float* __restrict__ h, const int* __restrict__ src,
                            const int* __restrict__ dst, const float* __restrict__ ew,
                            const float* __restrict__ dinv, float* __restrict__ agg) {
  long long t = (long long)blockIdx.x * blockDim.x + threadIdx.x;
  int e = (int)(t >> 4);
  if (e >= N_EDGES) return;
  int l = (int)(t & 15);
  int s = src[e], d = dst[e];
  float norm = dinv[s] * ew[e] * dinv[d];
  float4 v = ((const float4*)(h + (size_t)s * HID_C))[l];
  float* o = agg + (size_t)d * HID_C + l * 4;
  atomicAdd(o + 0, v.x * norm);
  atomicAdd(o + 1, v.y * norm);
  atomicAdd(o + 2, v.z * norm);
  atomicAdd(o + 3, v.w * norm);
}

// Self-loop term + bias + ReLU, written in place over the linear output h.
__global__ void gcn_finalize(const float* __restrict__ agg, float* __restrict__ h,
                             const float* __restrict__ dinv, const float* __restrict__ bias) {
  long long t = (long long)blockIdx.x * blockDim.x + threadIdx.x;
  int i = (int)(t >> 6);
  if (i >= N_NODES) return;
  int c = (int)(t & 63);
  float s = dinv[i];
  float v = agg[t] + h[t] * (s * s) + bias[c];
  h[t] = fmaxf(v, 0.0f);
}

// ---------------------------------------------------------------------------
// Global mean pool: per-graph channel sums + node counts via atomics.
// ---------------------------------------------------------------------------
__global__ void gcn_pool(const float* __restrict__ h, const int* __restrict__ batch,
                         float* __restrict__ sums, float* __restrict__ cnt) {
  long long t = (long long)blockIdx.x * blockDim.x + threadIdx.x;
  int i = (int)(t >> 4);
  if (i >= N_NODES) return;
  int l = (int)(t & 15);
  int g = batch[i];
  float4 v = ((const float4*)(h + (size_t)i * HID_C))[l];
  float* o = sums + (size_t)g * HID_C + l * 4;
  atomicAdd(o + 0, v.x);
  atomicAdd(o + 1, v.y);
  atomicAdd(o + 2, v.z);
  atomicAdd(o + 3, v.w);
  if (l == 0) atomicAdd(&cnt[g], 1.0f);
}

// Head: out[g,o] = (sums[g,:]/max(cnt,1)) . Wl[:,o] + bl[o]   (256x2 outputs)
__global__ void gcn_head(const float* __restrict__ sums, const float* __restrict__ cnt,
                         const float* __restrict__ Wl, const float* __restrict__ bl,
                         float* __restrict__ out) {
  int t = blockIdx.x * blockDim.x + threadIdx.x;
  if (t >= N_GRAPHS * OUT_C) return;
  int g = t / OUT_C, o = t % OUT_C;
  float inv = 1.0f / fmaxf(cnt[g], 1.0f);
  float acc = 0.0f;
  #pragma unroll
  for (int k = 0; k < HID_C; ++k)
    acc += sums[(size_t)g * HID_C + k] * Wl[k * OUT_C + o];
  out[t] = acc * inv + bl[o];
}

// ---------------------------------------------------------------------------
extern "C" void kernel_launch(void* const* d_in, const int* in_sizes, int n_in,
                              void* d_out, int out_size, void* d_ws, size_t ws_size,
                              hipStream_t stream) {
  const float* x  = (const float*)d_in[0];
  const int*   ei = (const int*)  d_in[1];
  const float* ea = (const float*)d_in[2];
  const int*   bt = (const int*)  d_in[3];
  const float* W1 = (const float*)d_in[4];
  const float* b1 = (const float*)d_in[5];
  const float* W2 = (const float*)d_in[6];
  const float* b2 = (const float*)d_in[7];
  const float* Wl = (const float*)d_in[8];
  const float* bl = (const float*)d_in[9];
  const int* src = ei;             // edge_index[0,:]
  const int* dst = ei + N_EDGES;   // edge_index[1,:]

  // Workspace layout (floats): dinv | h1 | h2 | agg | sums | cnt  (~38.7 MB)
  float* ws   = (float*)d_ws;
  float* dinv = ws;
  float* h1   = dinv + N_NODES;
  float* h2   = h1 + (size_t)N_NODES * HID_C;
  float* agg  = h2 + (size_t)N_NODES * HID_C;
  float* sums = agg + (size_t)N_NODES * HID_C;
  float* cnt  = sums + (size_t)N_GRAPHS * HID_C;

  const int Mtiles = N_NODES / 16;                    // 3125, exact
  const long long te = (long long)N_EDGES * 16;       // scatter threads
  const long long tn = (long long)N_NODES * 64;       // finalize threads
  const long long tp = (long long)N_NODES * 16;       // pool threads

  // Degrees -> dinv (shared by both convs)
  gcn_deg_init  <<<(N_NODES + 255) / 256, 256, 0, stream>>>(dinv);
  gcn_deg_accum <<<(N_EDGES + 255) / 256, 256, 0, stream>>>(dst, ea, dinv);
  gcn_deg_finish<<<(N_NODES + 255) / 256, 256, 0, stream>>>(dinv);

  // Conv1: lin (WMMA) -> propagate -> bias+ReLU
  gcn_gemm_wmma<IN_C><<<Mtiles, 128, 0, stream>>>(x, W1, h1, Mtiles);
  (void)hipMemsetAsync(agg, 0, (size_t)N_NODES * HID_C * sizeof(float), stream);
  gcn_scatter <<<(int)((te + 255) / 256), 256, 0, stream>>>(h1, src, dst, ea, dinv, agg);
  gcn_finalize<<<(int)((tn + 255) / 256), 256, 0, stream>>>(agg, h1, dinv, b1);

  // Conv2
  gcn_gemm_wmma<HID_C><<<Mtiles, 128, 0, stream>>>(h1, W2, h2, Mtiles);
  (void)hipMemsetAsync(agg, 0, (size_t)N_NODES * HID_C * sizeof(float), stream);
  gcn_scatter <<<(int)((te + 255) / 256), 256, 0, stream>>>(h2, src, dst, ea, dinv, agg);
  gcn_finalize<<<(int)((tn + 255) / 256), 256, 0, stream>>>(agg, h2, dinv, b2);

  // Mean pool + head
  (void)hipMemsetAsync(sums, 0, (size_t)(N_GRAPHS * HID_C + N_GRAPHS) * sizeof(float), stream);
  gcn_pool<<<(int)((tp + 255) / 256), 256, 0, stream>>>(h2, bt, sums, cnt);
  gcn_head<<<(N_GRAPHS * OUT_C + 127) / 128, 128, 0, stream>>>(sums, cnt, Wl, bl, (float*)d_out);
}